// MyNet_71768903516550
// MI455X (gfx1250) — compile-verified
//
#include <hip/hip_runtime.h>
#include <math.h>

#define NEG_SLOPE 0.2f
#define TB 256
#define KT 32          // K-chunk staged per LDS buffer
#define LA_S 36        // padded LDS stride for A (16B-aligned rows, conflict-free)
#define LB_S 136       // padded LDS stride for B (16B-aligned rows, half-wave banks disjoint)

typedef float v2f __attribute__((ext_vector_type(2)));
typedef float v8f __attribute__((ext_vector_type(8)));
typedef int   v4i __attribute__((ext_vector_type(4)));

#if __has_builtin(__builtin_amdgcn_global_load_async_to_lds_b128)
#define HAVE_ASYNC_LDS 1
typedef __attribute__((address_space(1))) v4i gv4i;
typedef __attribute__((address_space(3))) v4i lv4i;
static __device__ __forceinline__ void async_b128(const void* g, void* l) {
    __builtin_amdgcn_global_load_async_to_lds_b128((gv4i*)g, (lv4i*)l, 0, 0);
}
static __device__ __forceinline__ void wait_async() {
#if __has_builtin(__builtin_amdgcn_s_wait_asynccnt)
    __builtin_amdgcn_s_wait_asynccnt(0);
#else
    asm volatile("s_wait_asynccnt 0x0" ::: "memory");
#endif
}
#else
#define HAVE_ASYNC_LDS 0
#endif

static __device__ __forceinline__ float leaky(float a) {
    return a >= 0.f ? a : NEG_SLOPE * a;
}

// float atomic-max via monotone int encodings
static __device__ __forceinline__ void atomicMaxF(float* addr, float val) {
    if (val >= 0.f)
        atomicMax((int*)addr, __float_as_int(val));
    else
        atomicMin((unsigned int*)addr, (unsigned int)__float_as_int(val));
}

__global__ void fill_kernel(float* __restrict__ p, float v, int n) {
    int t = blockIdx.x * blockDim.x + threadIdx.x;
    if (t < n) p[t] = v;
}

// ---------------------------------------------------------------------------
// LDS-tiled GEMM: Hout[Nrows,Ncol] = X[Nrows,K] @ W[K,Ncol] + bias (fp32 WMMA)
// Block = 256 threads (8 waves) -> 32-row strip x full Ncol (<=128).
// Async (global->LDS) double-buffered staging issued BEFORE compute each
// iteration so V_WMMA work hides global latency. Requires K % 32 == 0.
// A frag (16x4): lanes 0-15 K=0,1 ; lanes 16-31 K=2,3.
// B frag (4x16): vgpr0 = row K{0|2}, vgpr1 = row K{1|3}, N = lane%16.
// C/D: vgpr r -> M=r (lanes 0-15) / M=8+r (lanes 16-31), N = lane%16.
// ---------------------------------------------------------------------------
__global__ void __launch_bounds__(TB)
gemm_wmma_lds(const float* __restrict__ X,
              const float* __restrict__ W,
              const float* __restrict__ bias,
              float* __restrict__ Hout,
              int Nrows, int K, int Ncol, int ntn) {
    __shared__ float lA[2][32][LA_S];
    __shared__ float lB[2][KT][LB_S];

    const int tid    = threadIdx.x;
    const int waveid = tid >> 5;
    const int lane   = tid & 31;
    const int lane16 = lane & 15;
    const int half   = lane >> 4;       // 0 | 1
    const int kofs   = half << 1;       // 0 | 2
    const int m0     = blockIdx.x * 32;
    const int NCPAD  = ntn << 4;        // 128 / 64 / 48
    const int ntiles = ntn * 2;         // tiles in 32 x NCPAD strip

    // wave-owned tiles (wave-uniform)
    const int t0 = waveid, t1 = waveid + 8;
    const bool has0 = t0 < ntiles, has1 = t1 < ntiles;
    const int mt0 = t0 / ntn, nt0 = t0 - mt0 * ntn;
    const int mt1 = has1 ? (t1 / ntn) : 0;
    const int nt1 = has1 ? (t1 - mt1 * ntn) : 0;

    // A staging map: thread -> (row, 4 cols), b128 along K
    const int arow = tid >> 3;              // 0..31
    const int acol = (tid & 7) << 2;        // 0..28 step 4
    int arow_g = m0 + arow;
    if (arow_g >= Nrows) arow_g = Nrows - 1;
    const float* __restrict__ aptr = X + (size_t)arow_g * K + acol;

    // zero-fill pad columns once (only layer 3: cols [40,48))
    if (Ncol < NCPAD) {
        const int padw = NCPAD - Ncol;
        for (int idx = tid; idx < 2 * KT * padw; idx += TB) {
            int b   = idx / (KT * padw);
            int rem = idx - b * KT * padw;
            int r   = rem / padw;
            int c   = Ncol + (rem - r * padw);
            lB[b][r][c] = 0.f;
        }
    }

    const int gpr = NCPAD >> 2;             // float4 groups per B row
    auto stage = [&](int b, int kb) {
#if HAVE_ASYNC_LDS
        async_b128(aptr + kb, &lA[b][arow][acol]);
#else
        *(float4*)&lA[b][arow][acol] = *(const float4*)(aptr + kb);
#endif
        for (int idx = tid; idx < KT * gpr; idx += TB) {
            int r = idx / gpr;
            int c = (idx - r * gpr) << 2;
            if (c < Ncol) {                 // Ncol % 4 == 0: group all-in or all-out
#if HAVE_ASYNC_LDS
                async_b128(W + (size_t)(kb + r) * Ncol + c, &lB[b][r][c]);
#else
                *(float4*)&lB[b][r][c] = *(const float4*)(W + (size_t)(kb + r) * Ncol + c);
#endif
            }
        }
    };

    v8f acc0 = {0.f,0.f,0.f,0.f,0.f,0.f,0.f,0.f};
    v8f acc1 = {0.f,0.f,0.f,0.f,0.f,0.f,0.f,0.f};

    stage(0, 0);
#if HAVE_ASYNC_LDS
    wait_async();
#endif
    __syncthreads();

    int buf = 0;
    for (int kb = 0; kb < K; kb += KT) {
        if (kb + KT < K) stage(buf ^ 1, kb + KT);   // prefetch next chunk (async)

        if (has0) {
            const int mrow = mt0 * 16 + lane16;
            const int ncc  = nt0 * 16 + lane16;
#pragma unroll
            for (int ks = 0; ks < KT; ks += 4) {
                v2f a, b;
                a.x = lA[buf][mrow][ks + kofs];
                a.y = lA[buf][mrow][ks + kofs + 1];
                b.x = lB[buf][ks + kofs][ncc];
                b.y = lB[buf][ks + kofs + 1][ncc];
                acc0 = __builtin_amdgcn_wmma_f32_16x16x4_f32(
                    false, a, false, b, (short)0, acc0, false, false);
            }
        }
        if (has1) {
            const int mrow = mt1 * 16 + lane16;
            const int ncc  = nt1 * 16 + lane16;
#pragma unroll
            for (int ks = 0; ks < KT; ks += 4) {
                v2f a, b;
                a.x = lA[buf][mrow][ks + kofs];
                a.y = lA[buf][mrow][ks + kofs + 1];
                b.x = lB[buf][ks + kofs][ncc];
                b.y = lB[buf][ks + kofs + 1][ncc];
                acc1 = __builtin_amdgcn_wmma_f32_16x16x4_f32(
                    false, a, false, b, (short)0, acc1, false, false);
            }
        }

#if HAVE_ASYNC_LDS
        wait_async();
#endif
        __syncthreads();
        buf ^= 1;
    }

    auto store_tile = [&](const v8f& acc, int mt, int nt) {
        int col = nt * 16 + lane16;
        if (col < Ncol) {
            float bv = bias[col];
            int mbase = m0 + mt * 16 + (half << 3);
#pragma unroll
            for (int r = 0; r < 8; ++r) {
                int row = mbase + r;
                if (row < Nrows)
                    Hout[(size_t)row * Ncol + col] = acc[r] + bv;
            }
        }
    };
    if (has0) store_tile(acc0, mt0, nt0);
    if (has1) store_tile(acc1, mt1, nt1);
}

// alpha[n,h] = sum_c h[n,h,c] * att[h,c]
__global__ void alpha_kernel(const float* __restrict__ h,
                             const float* __restrict__ att,
                             float* __restrict__ alpha,
                             int Nn, int Hh, int Cc) {
    int t = blockIdx.x * blockDim.x + threadIdx.x;
    if (t >= Nn * Hh) return;
    int n = t / Hh, hh = t - n * Hh;
    const float* hp = h + ((size_t)n * Hh + hh) * Cc;
    const float* ap = att + hh * Cc;
    float s = 0.f;
    for (int c = 0; c < Cc; ++c) s += hp[c] * ap[c];
    alpha[t] = s;
}

static __device__ __forceinline__ void edge_sd(const int* __restrict__ ei,
                                               int E, int e, int& s, int& d) {
    if (e < E) { s = ei[e]; d = ei[E + e]; }
    else       { s = e - E; d = e - E; }        // self loops appended
}

__global__ void edge_count_kernel(const int* __restrict__ ei, int E, int Etot,
                                  float* __restrict__ cnt) {
    int e = blockIdx.x * blockDim.x + threadIdx.x;
    if (e >= Etot) return;
    int s, d; edge_sd(ei, E, e, s, d);
    atomicAdd(&cnt[d], 1.0f);
}

__global__ void edge_amax_kernel(const int* __restrict__ ei, int E, int total, int Hh,
                                 const float* __restrict__ alpha,
                                 float* __restrict__ amax) {
    int t = blockIdx.x * blockDim.x + threadIdx.x;
    if (t >= total) return;
    int e = t / Hh, hh = t - e * Hh;
    int s, d; edge_sd(ei, E, e, s, d);
    float a = leaky(alpha[s * Hh + hh]);
    atomicMaxF(&amax[d * Hh + hh], a);
}

__global__ void edge_denom_kernel(const int* __restrict__ ei, int E, int total, int Hh,
                                  const float* __restrict__ alpha,
                                  const float* __restrict__ amax,
                                  float* __restrict__ denom) {
    int t = blockIdx.x * blockDim.x + threadIdx.x;
    if (t >= total) return;
    int e = t / Hh, hh = t - e * Hh;
    int s, d; edge_sd(ei, E, e, s, d);
    float a = leaky(alpha[s * Hh + hh]);
    float ex = expf(a - amax[d * Hh + hh]);
    atomicAdd(&denom[d * Hh + hh], ex);
}

__global__ void edge_scatter_kernel(const int* __restrict__ ei, int E, int total,
                                    int Hh, int Cc,
                                    const float* __restrict__ alpha,
                                    const float* __restrict__ amax,
                                    const float* __restrict__ denom,
                                    const float* __restrict__ hfeat,
                                    float* __restrict__ sbuf) {
    int t = blockIdx.x * blockDim.x + threadIdx.x;
    if (t >= total) return;
    int e = t / Hh, hh = t - e * Hh;
    int s, d; edge_sd(ei, E, e, s, d);
    float a = leaky(alpha[s * Hh + hh]);
    float w = expf(a - amax[d * Hh + hh]) / denom[d * Hh + hh];
    const float* hs = hfeat + ((size_t)s * Hh + hh) * Cc;
    float* sd = sbuf + ((size_t)d * Hh + hh) * Cc;
    for (int c = 0; c < Cc; c += 4) {           // Cc in {16,8,40}: multiple of 4
        float4 hv = *(const float4*)(hs + c);
        atomicAdd(&sd[c + 0], hv.x * w);
        atomicAdd(&sd[c + 1], hv.y * w);
        atomicAdd(&sd[c + 2], hv.z * w);
        atomicAdd(&sd[c + 3], hv.w * w);
    }
}

__global__ void finalize_kernel(const float* __restrict__ sbuf,
                                const float* __restrict__ cnt,
                                const float* __restrict__ bias,
                                float* __restrict__ out,
                                int Nn, int HC, int do_leaky) {
    int t = blockIdx.x * blockDim.x + threadIdx.x;
    if (t >= Nn * HC) return;
    int n = t / HC, j = t - n * HC;
    float v = sbuf[t] / cnt[n] + bias[j];
    if (do_leaky) v = leaky(v);
    out[t] = v;
}

// layer-3 finalize + row-wise log_softmax (C <= 64)
__global__ void finalize_lsm_kernel(const float* __restrict__ sbuf,
                                    const float* __restrict__ cnt,
                                    const float* __restrict__ bias,
                                    float* __restrict__ out,
                                    int Nn, int Cc) {
    int n = blockIdx.x * blockDim.x + threadIdx.x;
    if (n >= Nn) return;
    float inv = 1.0f / cnt[n];
    float v[64];
    float m = -INFINITY;
    for (int c = 0; c < Cc; ++c) {
        v[c] = sbuf[(size_t)n * Cc + c] * inv + bias[c];
        m = fmaxf(m, v[c]);
    }
    float sum = 0.f;
    for (int c = 0; c < Cc; ++c) sum += expf(v[c] - m);
    float l = logf(sum);
    for (int c = 0; c < Cc; ++c)
        out[(size_t)n * Cc + c] = v[c] - m - l;
}

static inline int cdiv(int a, int b) { return (a + b - 1) / b; }

extern "C" void kernel_launch(void* const* d_in, const int* in_sizes, int n_in,
                              void* d_out, int out_size, void* d_ws, size_t ws_size,
                              hipStream_t stream) {
    const int F_IN = 512;
    const int Nn = in_sizes[0] / F_IN;     // 50000
    const int E  = in_sizes[1] / 2;        // 800000
    const int Etot = E + Nn;

    const float* x   = (const float*)d_in[0];
    const int*   ei  = (const int*)d_in[1];
    const float* W1  = (const float*)d_in[2];
    const float* lb1 = (const float*)d_in[3];
    const float* at1 = (const float*)d_in[4];
    const float* b1  = (const float*)d_in[5];
    const float* W2  = (const float*)d_in[6];
    const float* lb2 = (const float*)d_in[7];
    const float* at2 = (const float*)d_in[8];
    const float* b2  = (const float*)d_in[9];
    const float* W3  = (const float*)d_in[10];
    const float* lb3 = (const float*)d_in[11];
    const float* at3 = (const float*)d_in[12];
    const float* b3  = (const float*)d_in[13];
    float* out = (float*)d_out;

    // workspace layout (floats): h | sbuf | alpha | amax | denom | cnt | xbuf
    float* ws    = (float*)d_ws;
    float* hbuf  = ws;
    float* sbuf  = hbuf  + (size_t)Nn * 128;
    float* alph  = sbuf  + (size_t)Nn * 128;
    float* amax  = alph  + (size_t)Nn * 8;
    float* denom = amax  + (size_t)Nn * 8;
    float* cnt   = denom + (size_t)Nn * 8;
    float* xbuf  = cnt   + (size_t)Nn;

    // in-degree counts (same graph for all layers)
    fill_kernel<<<cdiv(Nn, TB), TB, 0, stream>>>(cnt, 0.f, Nn);
    edge_count_kernel<<<cdiv(Etot, TB), TB, 0, stream>>>(ei, E, Etot, cnt);

    struct Layer {
        const float *Xin, *Wm, *lb, *att, *bias;
        int K, Hh, Cc;
        float* act_out;   // nullptr => final layer (log_softmax to d_out)
    };
    Layer layers[3] = {
        { x,    W1, lb1, at1, b1, 512, 8, 16, xbuf },
        { xbuf, W2, lb2, at2, b2, 128, 8,  8, xbuf },
        { xbuf, W3, lb3, at3, b3,  64, 1, 40, nullptr },
    };

    for (int li = 0; li < 3; ++li) {
        const Layer& L = layers[li];
        const int HC = L.Hh * L.Cc;
        const int ntn = cdiv(HC, 16);

        // 1) h = X @ W + lb   (fp32 WMMA, async LDS double-buffered)
        gemm_wmma_lds<<<cdiv(Nn, 32), TB, 0, stream>>>(
            L.Xin, L.Wm, L.lb, hbuf, Nn, L.K, HC, ntn);

        // 2) alpha = sum_c h*att
        alpha_kernel<<<cdiv(Nn * L.Hh, TB), TB, 0, stream>>>(
            hbuf, L.att, alph, Nn, L.Hh, L.Cc);

        // 3) segment softmax + scatter
        fill_kernel<<<cdiv(Nn * L.Hh, TB), TB, 0, stream>>>(amax, -INFINITY, Nn * L.Hh);
        fill_kernel<<<cdiv(Nn * L.Hh, TB), TB, 0, stream>>>(denom, 0.f, Nn * L.Hh);
        fill_kernel<<<cdiv(Nn * HC, TB), TB, 0, stream>>>(sbuf, 0.f, Nn * HC);

        const int etot_h = Etot * L.Hh;
        edge_amax_kernel<<<cdiv(etot_h, TB), TB, 0, stream>>>(
            ei, E, etot_h, L.Hh, alph, amax);
        edge_denom_kernel<<<cdiv(etot_h, TB), TB, 0, stream>>>(
            ei, E, etot_h, L.Hh, alph, amax, denom);
        edge_scatter_kernel<<<cdiv(etot_h, TB), TB, 0, stream>>>(
            ei, E, etot_h, L.Hh, L.Cc, alph, amax, denom, hbuf, sbuf);

        // 4) mean aggregation + bias (+ leaky / log_softmax)
        if (L.act_out) {
            finalize_kernel<<<cdiv(Nn * HC, TB), TB, 0, stream>>>(
                sbuf, cnt, L.bias, L.act_out, Nn, HC, 1);
        } else {
            finalize_lsm_kernel<<<cdiv(Nn, TB), TB, 0, stream>>>(
                sbuf, cnt, L.bias, out, Nn, L.Cc);
        }
    }
}